// LoRM_5480378270401
// MI455X (gfx1250) — compile-verified
//
#include <hip/hip_runtime.h>
#include <math.h>

typedef __attribute__((ext_vector_type(2))) float v2f;
typedef __attribute__((ext_vector_type(8))) float v8f;

// ---------------------------------------------------------------------------
// mask[b,n] = (bilinear_align_corners(where(cam==255,0,cam), 48x48) > 0)
// ---------------------------------------------------------------------------
__global__ __launch_bounds__(256)
void mask_kernel(const float* __restrict__ cam, float* __restrict__ mask, int B) {
    int idx = blockIdx.x * 256 + threadIdx.x;
    if (idx >= B * 2304) return;
    int b = idx / 2304, p = idx % 2304;
    int y = p / 48, x = p % 48;
    const float* cb = cam + (long long)b * 384 * 384;
    const float S = 383.0f / 47.0f;
    float ys = (float)y * S, xs = (float)x * S;
    int y0 = (int)floorf(ys), x0 = (int)floorf(xs);
    int y1 = (y0 + 1 < 383) ? y0 + 1 : 383;
    int x1 = (x0 + 1 < 383) ? x0 + 1 : 383;
    float wy = ys - (float)y0, wx = xs - (float)x0;
    float v00 = cb[y0 * 384 + x0]; v00 = (v00 == 255.0f) ? 0.0f : v00;
    float v01 = cb[y0 * 384 + x1]; v01 = (v01 == 255.0f) ? 0.0f : v01;
    float v10 = cb[y1 * 384 + x0]; v10 = (v10 == 255.0f) ? 0.0f : v10;
    float v11 = cb[y1 * 384 + x1]; v11 = (v11 == 255.0f) ? 0.0f : v11;
    float top = (1.0f - wx) * v00 + wx * v01;
    float bot = (1.0f - wx) * v10 + wx * v11;
    float v = (1.0f - wy) * top + wy * bot;
    mask[idx] = (v > 0.0f) ? 1.0f : 0.0f;
}

// ---------------------------------------------------------------------------
// WMMA f32 GEMM, row-row form: D[i,j] = sum_k A[i,k] * B[j,k]
//   TRANSA=true : A stored [K rows x M cols], lda = row stride (coalesced)
//   FUSE_LOSS   : no D store; r=(x - gamma*D)^2, block partial sum / nnz count
// Block: 128 threads = 4 waves; block tile 128x128; wave tile 64x64 (4x4 WMMA
// accumulators). K staged in 16-wide chunks; double-buffered LDS with register
// prefetch so global loads overlap the v_wmma_f32_16x16x4_f32 stream.
// ---------------------------------------------------------------------------
template<bool TRANSA, bool FUSE_LOSS>
__global__ __launch_bounds__(128)
void gemm_rr(const float* __restrict__ A, const float* __restrict__ Bm,
             float* __restrict__ D, const float* __restrict__ bias,
             int M, int Nn, int K, int lda, int ldb,
             long long sA, long long sB, long long sD,
             const float* __restrict__ X, long long sX, int ldX,
             const float* __restrict__ gamma,
             float* __restrict__ psum, unsigned* __restrict__ pcnt)
{
    constexpr int LDT = 20;                 // LDS row stride (dwords)
    __shared__ float As[2][128 * LDT];
    __shared__ float Bs[2][128 * LDT];

    const int z = blockIdx.z;
    const float* Ab = A + (long long)z * sA;
    const float* Bb = Bm + (long long)z * sB;

    const int i0 = blockIdx.y * 128;
    const int j0 = blockIdx.x * 128;
    const int tid  = threadIdx.x;
    const int lane = tid & 31;
    const int wave = tid >> 5;
    const int wm = (wave >> 1) * 64;
    const int wn = (wave & 1) * 64;
    const int fr = lane & 15;               // fragment row within 16
    const int kb = (lane >> 4) * 2;         // K sub-offset per ISA A/B layout

    v8f acc[4][4] = {};

    float4 pa4[4];
    float  pas[16];
    float4 pb4[4];

    auto load_a = [&](int kbase) {
        if (TRANSA) {
#pragma unroll
            for (int q = 0; q < 16; ++q)
                pas[q] = Ab[(long long)(kbase + q) * lda + (i0 + tid)];
        } else {
#pragma unroll
            for (int q = 0; q < 4; ++q) {
                int e = tid * 4 + q * 512;
                pa4[q] = *(const float4*)(Ab + (long long)(i0 + (e >> 4)) * lda + kbase + (e & 15));
            }
        }
    };
    auto load_b = [&](int kbase) {
#pragma unroll
        for (int q = 0; q < 4; ++q) {
            int e = tid * 4 + q * 512;
            pb4[q] = *(const float4*)(Bb + (long long)(j0 + (e >> 4)) * ldb + kbase + (e & 15));
        }
    };
    auto store_ab = [&](int buf) {
        if (TRANSA) {
#pragma unroll
            for (int q = 0; q < 16; ++q)
                As[buf][tid * LDT + q] = pas[q];
        } else {
#pragma unroll
            for (int q = 0; q < 4; ++q) {
                int e = tid * 4 + q * 512;
                *(float4*)(&As[buf][(e >> 4) * LDT + (e & 15)]) = pa4[q];
            }
        }
#pragma unroll
        for (int q = 0; q < 4; ++q) {
            int e = tid * 4 + q * 512;
            *(float4*)(&Bs[buf][(e >> 4) * LDT + (e & 15)]) = pb4[q];
        }
    };

    const int nk = K >> 4;
    load_a(0);
    load_b(0);
    store_ab(0);
    __syncthreads();

    for (int kc = 0; kc < nk; ++kc) {
        const int cur = kc & 1;
        const bool more = (kc + 1 < nk);
        if (more) { load_a((kc + 1) << 4); load_b((kc + 1) << 4); }

#pragma unroll
        for (int kk = 0; kk < 16; kk += 4) {
            v2f af[4], bf[4];
#pragma unroll
            for (int t = 0; t < 4; ++t) {
                af[t] = *(const v2f*)&As[cur][(wm + t * 16 + fr) * LDT + kk + kb];
                bf[t] = *(const v2f*)&Bs[cur][(wn + t * 16 + fr) * LDT + kk + kb];
            }
#pragma unroll
            for (int ti = 0; ti < 4; ++ti)
#pragma unroll
            for (int tj = 0; tj < 4; ++tj)
                acc[ti][tj] = __builtin_amdgcn_wmma_f32_16x16x4_f32(
                    false, af[ti], false, bf[tj], (short)0, acc[ti][tj], false, false);
        }

        if (more) {
            store_ab(cur ^ 1);
            __syncthreads();
        }
    }

    // D layout (16x16 f32): VGPR r -> M = r + 8*(lane>=16), N = lane&15
    const int mhi = (lane >> 4) * 8;
    const int nlo = lane & 15;

    if (!FUSE_LOSS) {
        float* Db = D + (long long)z * sD;
#pragma unroll
        for (int ti = 0; ti < 4; ++ti)
#pragma unroll
        for (int tj = 0; tj < 4; ++tj)
#pragma unroll
        for (int r = 0; r < 8; ++r) {
            int grow = i0 + wm + ti * 16 + mhi + r;
            int gcol = j0 + wn + tj * 16 + nlo;
            float v = acc[ti][tj][r];
            if (bias) v += bias[gcol];
            Db[(long long)grow * Nn + gcol] = v;
        }
    } else {
        const float* Xb = X + (long long)z * sX;
        const float g = gamma[0];
        float lsum = 0.0f;
        unsigned lcnt = 0;
#pragma unroll
        for (int ti = 0; ti < 4; ++ti)
#pragma unroll
        for (int tj = 0; tj < 4; ++tj)
#pragma unroll
        for (int r = 0; r < 8; ++r) {
            int grow = i0 + wm + ti * 16 + mhi + r;   // pixel n
            int gcol = j0 + wn + tj * 16 + nlo;       // channel c
            float nx = g * acc[ti][tj][r];
            float xv = Xb[(long long)gcol * ldX + grow];
            float d  = xv - nx;
            float rq = d * d;
            lsum += rq;
            lcnt += (rq != 0.0f) ? 1u : 0u;
        }
        // deterministic block reduction (reuse LDS; all waves past last sync)
        __syncthreads();
        float*    red_s = &As[0][0];
        unsigned* red_c = (unsigned*)&Bs[0][0];
        red_s[tid] = lsum;
        red_c[tid] = lcnt;
        __syncthreads();
        for (int s = 64; s > 0; s >>= 1) {
            if (tid < s) { red_s[tid] += red_s[tid + s]; red_c[tid] += red_c[tid + s]; }
            __syncthreads();
        }
        if (tid == 0) {
            int blin = ((int)blockIdx.z * (int)gridDim.y + (int)blockIdx.y) * (int)gridDim.x + (int)blockIdx.x;
            psum[blin] = red_s[0];
            pcnt[blin] = red_c[0];
        }
    }
}

// ---------------------------------------------------------------------------
// out[row] = sqrt(sum_k A[row,k]^2) ; one block per row
// ---------------------------------------------------------------------------
__global__ __launch_bounds__(256)
void rownorm_kernel(const float* __restrict__ A, float* __restrict__ out, int K) {
    __shared__ float sm[256];
    const float* row = A + (long long)blockIdx.x * K;
    float s = 0.0f;
    for (int c = threadIdx.x; c < K; c += 256) { float v = row[c]; s += v * v; }
    sm[threadIdx.x] = s;
    __syncthreads();
    for (int w = 128; w > 0; w >>= 1) {
        if (threadIdx.x < w) sm[threadIdx.x] += sm[threadIdx.x + w];
        __syncthreads();
    }
    if (threadIdx.x == 0) out[blockIdx.x] = sqrtf(sm[0]);
}

// ---------------------------------------------------------------------------
// In-place: row <- mask[row] * softmax_m( row[m] / (qn[row]*kn[b,m]) )
// One block per (b,n) row of E.
// ---------------------------------------------------------------------------
__global__ __launch_bounds__(256)
void softmax_mask_kernel(float* __restrict__ E, const float* __restrict__ qn,
                         const float* __restrict__ kn, const float* __restrict__ mask,
                         int N) {
    __shared__ float sm[256];
    const long long rg = blockIdx.x;
    const int b = (int)(rg / N);
    float* row = E + rg * (long long)N;
    const float* knb = kn + (long long)b * N;
    const float q = qn[rg];
    const int tid = threadIdx.x;

    float mx = -INFINITY;
    for (int m = tid; m < N; m += 256) {
        float l = row[m] / (q * knb[m]);
        row[m] = l;
        mx = fmaxf(mx, l);
    }
    sm[tid] = mx; __syncthreads();
    for (int w = 128; w > 0; w >>= 1) { if (tid < w) sm[tid] = fmaxf(sm[tid], sm[tid + w]); __syncthreads(); }
    mx = sm[0]; __syncthreads();

    float s = 0.0f;
    for (int m = tid; m < N; m += 256) {
        float e = expf(row[m] - mx);
        row[m] = e;
        s += e;
    }
    sm[tid] = s; __syncthreads();
    for (int w = 128; w > 0; w >>= 1) { if (tid < w) sm[tid] += sm[tid + w]; __syncthreads(); }
    s = sm[0];

    const float scale = mask[rg] / s;
    for (int m = tid; m < N; m += 256) row[m] *= scale;
}

// ---------------------------------------------------------------------------
// Serial deterministic finalize: loss = sum(partials) / count
// ---------------------------------------------------------------------------
__global__ void finalize_kernel(const float* __restrict__ psum,
                                const unsigned* __restrict__ pcnt,
                                int n, float* __restrict__ out) {
    if (threadIdx.x == 0 && blockIdx.x == 0) {
        double s = 0.0;
        unsigned long long c = 0;
        for (int i = 0; i < n; ++i) { s += (double)psum[i]; c += pcnt[i]; }
        out[0] = (float)(s / (double)c);
    }
}

// ---------------------------------------------------------------------------
extern "C" void kernel_launch(void* const* d_in, const int* in_sizes, int n_in,
                              void* d_out, int out_size, void* d_ws, size_t ws_size,
                              hipStream_t stream) {
    const float* x     = (const float*)d_in[0];   // [4,2048,48,48]
    const float* cam   = (const float*)d_in[1];   // [4,384,384]
    const float* Wq    = (const float*)d_in[2];   // [2048,2048]
    const float* bq    = (const float*)d_in[3];   // [2048]
    const float* Wk    = (const float*)d_in[4];   // [2048,2048]
    const float* bk    = (const float*)d_in[5];   // [2048]
    const float* gamma = (const float*)d_in[6];   // [1]

    const int B = 4, C = 2048, N = 2304;

    char* ws = (char*)d_ws;
    size_t off = 0;
    auto take = [&](size_t bytes) -> char* {
        char* p = ws + off;
        off = (off + bytes + 255) & ~(size_t)255;
        return p;
    };
    float*    Q    = (float*)take((size_t)B * N * C * sizeof(float));  // [B,N,C]
    float*    Kt   = (float*)take((size_t)B * N * C * sizeof(float));  // [B,N,C] (k transposed)
    float*    E    = (float*)take((size_t)B * N * N * sizeof(float));  // energy -> atten in place
    float*    qn   = (float*)take((size_t)B * N * sizeof(float));
    float*    kn   = (float*)take((size_t)B * N * sizeof(float));
    float*    mask = (float*)take((size_t)B * N * sizeof(float));
    const int NB3  = (C / 128) * (N / 128) * B;                        // 1152 partials
    float*    psum = (float*)take((size_t)NB3 * sizeof(float));
    unsigned* pcnt = (unsigned*)take((size_t)NB3 * sizeof(unsigned));
    (void)ws_size; (void)in_sizes; (void)n_in; (void)out_size;

    // 1) query mask from cam
    mask_kernel<<<dim3((B * N + 255) / 256), dim3(256), 0, stream>>>(cam, mask, B);

    // 2) Q[b,n,o] = sum_c x[b,c,n] * Wq[o,c] + bq[o]   (TRANSA reads x directly)
    {
        dim3 g(C / 128, N / 128, B), blk(128);
        gemm_rr<true, false><<<g, blk, 0, stream>>>(
            x, Wq, Q, bq, N, C, C, /*lda=*/N, /*ldb=*/C,
            (long long)C * N, 0LL, (long long)N * C,
            nullptr, 0LL, 0, nullptr, nullptr, nullptr);
        gemm_rr<true, false><<<g, blk, 0, stream>>>(
            x, Wk, Kt, bk, N, C, C, /*lda=*/N, /*ldb=*/C,
            (long long)C * N, 0LL, (long long)N * C,
            nullptr, 0LL, 0, nullptr, nullptr, nullptr);
    }

    // 3) row norms of Q and Kt
    rownorm_kernel<<<dim3(B * N), dim3(256), 0, stream>>>(Q, qn, C);
    rownorm_kernel<<<dim3(B * N), dim3(256), 0, stream>>>(Kt, kn, C);

    // 4) energy[b,n,m] = sum_c Q[b,n,c] * Kt[b,m,c]
    {
        dim3 g(N / 128, N / 128, B), blk(128);
        gemm_rr<false, false><<<g, blk, 0, stream>>>(
            Q, Kt, E, nullptr, N, N, C, /*lda=*/C, /*ldb=*/C,
            (long long)N * C, (long long)N * C, (long long)N * N,
            nullptr, 0LL, 0, nullptr, nullptr, nullptr);
    }

    // 5) scaled softmax + query-row mask, in place
    softmax_mask_kernel<<<dim3(B * N), dim3(256), 0, stream>>>(E, qn, kn, mask, N);

    // 6) new_x[b,c,n] = sum_m atten[b,n,m] * x[b,c,m]; fused r=(x - g*new_x)^2 reduce
    {
        dim3 g(C / 128, N / 128, B), blk(128);
        gemm_rr<false, true><<<g, blk, 0, stream>>>(
            E, x, nullptr, nullptr, N, C, N, /*lda=*/N, /*ldb=*/N,
            (long long)N * N, (long long)C * N, 0LL,
            x, (long long)C * N, /*ldX=*/N, gamma, psum, pcnt);
    }

    // 7) loss = sum(r) / count_nonzero(r)
    finalize_kernel<<<dim3(1), dim3(1), 0, stream>>>(psum, pcnt, NB3, (float*)d_out);
}